// SoftDecisionML10_5_1726576857965
// MI455X (gfx1250) — compile-verified
//
#include <hip/hip_runtime.h>

// ---------------------------------------------------------------------------
// SoftDecisionML10: nearest-codeword decode.
// Reduces to argmax_k <signal, codebook_k> because ||c_k||^2 == 10 for all k
// and softmax(-sqrt(.)) is monotone.  Memory-bound (~336MB @ 23.3TB/s ~ 14us),
// so dot products go through V_WMMA_F32_16X16X4_F32 to keep VALU off the
// critical path, and signal/output use non-temporal hints (stream > 192MB L2).
// K=10 padding is done branchlessly (clamped address + cndmask zeroing) so the
// load stream has no exec divergence and no per-iteration full loadcnt stalls.
// ---------------------------------------------------------------------------

typedef __attribute__((ext_vector_type(2))) float v2f;
typedef __attribute__((ext_vector_type(4))) float v4f;
typedef __attribute__((ext_vector_type(8))) float v8f;

#define WAVES_PER_BLOCK 8
#define SCORE_STRIDE    36   // floats per point-row in LDS: 16B-aligned rows,
                             // conflict-free scatter (halves land 32 banks apart)

__global__ __launch_bounds__(256) void sdml10_wmma_kernel(
    const float* __restrict__ sig,   // [P, 10]
    const float* __restrict__ cb,    // [32, 10]
    float* __restrict__ out)         // [P, 10]
{
  __shared__ float scores[WAVES_PER_BLOCK * 32 * SCORE_STRIDE];

  const int lane = threadIdx.x & 31;
  const int wave = threadIdx.x >> 5;
  const int hl   = lane >> 4;     // wave half (selects K pair per ISA layout)
  const int lm   = lane & 15;     // M/N index within a 16-wide tile
  const long long wavePoint0 = (long long)blockIdx.x * 256 + wave * 32;

  // ---- B operands: codebook^T, 2 tiles (codewords 0-15 / 16-31) x 3 K-steps.
  // f32 4x16 B layout: lane l holds B[k0+v][n], n = l%16, k0 = 4s + 2*(l/16).
  // Branchless K padding: clamp address to a valid pair, zero via selects.
  v2f b[2][3];
#pragma unroll
  for (int t = 0; t < 2; ++t) {
    const int n = 16 * t + lm;
#pragma unroll
    for (int s = 0; s < 3; ++s) {
      const int  k0    = 4 * s + 2 * hl;            // 0,2,4,6,8,10
      const int  kc    = (k0 <= 8) ? k0 : 8;        // always-valid address
      const bool valid = (k0 < 10);
      v2f raw = *(const v2f*)(cb + n * 10 + kc);    // tiny table, cached
      v2f v; v.x = valid ? raw.x : 0.f; v.y = valid ? raw.y : 0.f;
      b[t][s] = v;
    }
  }

  // ---- Load A (two groups of 16 points) and run WMMAs.
  // f32 16x4 A layout: lane l holds A[m][k0+v], m = l%16, k0 = 2*(l/16) (+4s).
  v8f acc[2][2];
#pragma unroll
  for (int g = 0; g < 2; ++g) {
    const long long p = wavePoint0 + 16 * g + lm;
    const float* sp = sig + p * 10;
    v2f a[3];
#pragma unroll
    for (int s = 0; s < 3; ++s) {
      const int  k0    = 4 * s + 2 * hl;
      const int  kc    = (k0 <= 8) ? k0 : 8;
      const bool valid = (k0 < 10);
      v2f raw = __builtin_nontemporal_load((const v2f*)(sp + kc));
      v2f v; v.x = valid ? raw.x : 0.f; v.y = valid ? raw.y : 0.f;
      a[s] = v;
    }
#pragma unroll
    for (int t = 0; t < 2; ++t) {
      v8f c = {0.f, 0.f, 0.f, 0.f, 0.f, 0.f, 0.f, 0.f};
#pragma unroll
      for (int s = 0; s < 3; ++s) {
        // D = A(16x4,f32) x B(4x16,f32) + C ; chained over K
        c = __builtin_amdgcn_wmma_f32_16x16x4_f32(
                false, a[s], false, b[t][s], (short)0, c, false, false);
      }
      acc[g][t] = c;
    }
  }

  // ---- Scatter 32x32 score tile to LDS.
  // D layout: VGPR r, lanes 0-15 -> M=r, N=lane ; lanes 16-31 -> M=r+8.
  float* sbase = scores + wave * 32 * SCORE_STRIDE;
#pragma unroll
  for (int g = 0; g < 2; ++g)
#pragma unroll
    for (int t = 0; t < 2; ++t)
#pragma unroll
      for (int r = 0; r < 8; ++r) {
        const int m = 16 * g + 8 * hl + r;
        const int cidx = 16 * t + lm;
        sbase[m * SCORE_STRIDE + cidx] = acc[g][t][r];
      }

  __syncthreads();   // wave-private regions; barrier just orders DS ops

  // ---- One point per lane: argmax over 32 scores (first-max wins, matching
  // jnp.argmax tie-breaking), then emit codebook[best].
  const float* row = sbase + lane * SCORE_STRIDE;
  float best = -3.402823466e38f;
  int bi = 0;
#pragma unroll
  for (int q = 0; q < 8; ++q) {
    v4f v = *(const v4f*)(row + 4 * q);
#pragma unroll
    for (int j = 0; j < 4; ++j) {
      const float s = v[j];
      if (s > best) { best = s; bi = 4 * q + j; }
    }
  }

  const long long p = wavePoint0 + lane;
  const float* crow = cb + bi * 10;
  float* orow = out + p * 10;
#pragma unroll
  for (int j = 0; j < 5; ++j) {
    v2f v = *(const v2f*)(crow + 2 * j);
    __builtin_nontemporal_store(v, (v2f*)(orow + 2 * j));
  }
}

// Scalar tail for point counts not divisible by 256 (not hit for this shape,
// but keeps the launch correct for any P).
__global__ void sdml10_tail_kernel(const float* __restrict__ sig,
                                   const float* __restrict__ cb,
                                   float* __restrict__ out,
                                   long long start, long long total)
{
  long long p = start + blockIdx.x * blockDim.x + threadIdx.x;
  if (p >= total) return;
  float x[10];
#pragma unroll
  for (int d = 0; d < 10; ++d) x[d] = sig[p * 10 + d];
  float best = -3.402823466e38f; int bi = 0;
  for (int k = 0; k < 32; ++k) {
    float dot = 0.f;
#pragma unroll
    for (int d = 0; d < 10; ++d) dot += x[d] * cb[k * 10 + d];
    if (dot > best) { best = dot; bi = k; }
  }
#pragma unroll
  for (int d = 0; d < 10; ++d) out[p * 10 + d] = cb[bi * 10 + d];
}

extern "C" void kernel_launch(void* const* d_in, const int* in_sizes, int n_in,
                              void* d_out, int out_size, void* d_ws, size_t ws_size,
                              hipStream_t stream) {
  const float* sig = (const float*)d_in[0];   // [B*N, 10] fp32
  const float* cb  = (const float*)d_in[1];   // [32, 10]  fp32
  float* out = (float*)d_out;                 // [B*N, 10] fp32

  const long long npts  = (long long)in_sizes[0] / 10;   // 4,194,304
  const long long nfull = npts / 256;                    // 16,384 blocks
  if (nfull > 0) {
    sdml10_wmma_kernel<<<(int)nfull, 256, 0, stream>>>(sig, cb, out);
  }
  const long long rem = npts - nfull * 256;
  if (rem > 0) {
    const int tb = 256;
    const int tg = (int)((rem + tb - 1) / tb);
    sdml10_tail_kernel<<<tg, tb, 0, stream>>>(sig, cb, out, nfull * 256, npts);
  }
}